// SoftmaxOverNBest_74869869904422
// MI455X (gfx1250) — compile-verified
//
#include <hip/hip_runtime.h>
#include <hip/hip_bf16.h>

// Segmented softmax over ragged n-best groups.
// Memory-bound (~404 MB total traffic -> ~17us floor at 23.3 TB/s HBM):
// pure streaming pipeline with wide b128 VMEM, direct global->LDS async
// staging (gfx1250 ASYNCcnt path) overlapped with the segment-offset scan,
// wave32 shuffle reductions, hardware exp.

#define SEGS_PER_BLOCK 8
#define BLOCK_THREADS  256
#define LDS_CAP        2048   // floats staged per block (8 KB LDS)

#ifndef __has_builtin
#define __has_builtin(x) 0
#endif

#if defined(__HIP_DEVICE_COMPILE__) && __has_builtin(__builtin_amdgcn_global_load_async_to_lds_b128)
#define HAVE_ASYNC_LDS_LOAD 1
#else
#define HAVE_ASYNC_LDS_LOAD 0
#endif

typedef __attribute__((ext_vector_type(4))) int v4i;
typedef __attribute__((address_space(1))) v4i* gv4i_p;   // global int4*
typedef __attribute__((address_space(3))) v4i* lv4i_p;   // LDS int4*

__device__ __forceinline__ void wait_asynccnt0() {
#if defined(__HIP_DEVICE_COMPILE__)
#if __has_builtin(__builtin_amdgcn_s_wait_asynccnt)
    __builtin_amdgcn_s_wait_asynccnt(0);
#else
    asm volatile("s_wait_asynccnt 0x0" ::: "memory");
#endif
#endif
}

// ---------------------------------------------------------------------------
// Kernel A: per-softmax-block element counts (sum of 8 segment counts).
// Interior blocks: two b128 loads, branch-free. Tail block: scalar loop.
// ---------------------------------------------------------------------------
__global__ void nbest_block_sums(const int* __restrict__ nbest, int num_seg,
                                 int* __restrict__ bsum, int ncb) {
    int i = blockIdx.x * blockDim.x + threadIdx.x;
    if (i >= ncb) return;
    int base = i * SEGS_PER_BLOCK;
    int s;
    if (base + SEGS_PER_BLOCK <= num_seg) {
        const int4* p = (const int4*)(nbest + base);   // 32B-aligned
        int4 a = p[0];
        int4 c = p[1];
        s = a.x + a.y + a.z + a.w + c.x + c.y + c.z + c.w;
    } else {
        s = 0;
        for (int j = 0; j < SEGS_PER_BLOCK; ++j) {
            int seg = base + j;
            if (seg < num_seg) s += nbest[seg];
        }
    }
    bsum[i] = s;
}

// ---------------------------------------------------------------------------
// Kernel B: in-place exclusive scan of the block sums; additionally writes
// the grand total at data[n] so consumers can get any block's element count
// as bases[b+1]-bases[b] without waiting on the per-segment count scan.
// ---------------------------------------------------------------------------
__global__ void exclusive_scan_inplace(int* __restrict__ data, int n) {
    __shared__ int tile[1024];
    __shared__ int carry;
    int tid = threadIdx.x;
    if (tid == 0) carry = 0;
    __syncthreads();
    for (int base = 0; base < n; base += 1024) {
        int i = base + tid;
        int v = (i < n) ? data[i] : 0;
        tile[tid] = v;
        __syncthreads();
        for (int off = 1; off < 1024; off <<= 1) {
            int t = (tid >= off) ? tile[tid - off] : 0;
            __syncthreads();
            tile[tid] += t;
            __syncthreads();
        }
        int incl = tile[tid];
        int c = carry;                 // read before thread 0 updates it
        if (i < n) data[i] = c + incl - v;   // exclusive prefix
        __syncthreads();
        if (tid == 0) carry += tile[1023];
        __syncthreads();
    }
    if (tid == 0) data[n] = carry;     // grand total sentinel
}

// ---------------------------------------------------------------------------
// Kernel C: segmented softmax. One wave32 per segment, 8 segments per block.
// Staging copy issued immediately (async global->LDS when available),
// overlapped with the per-segment offset scan; shuffle butterflies for wave
// reductions; v_exp_f32 for exp; b128 store-out.
// ---------------------------------------------------------------------------
__global__ void __launch_bounds__(BLOCK_THREADS)
segmented_softmax(const float* __restrict__ scores,
                  const int* __restrict__ nbest,
                  float* __restrict__ out,
                  const int* __restrict__ block_bases,
                  int num_seg) {
    __shared__ __align__(16) float vals[LDS_CAP];
    __shared__ int s_cnt[SEGS_PER_BLOCK];
    __shared__ int s_start[SEGS_PER_BLOCK];

    const int b = blockIdx.x;
    const int tid = threadIdx.x;
    const int gbase = block_bases[b];              // uniform -> scalar load
    const int total = block_bases[b + 1] - gbase;  // element count, no scan needed

    const bool in_lds = (total <= LDS_CAP);
    const bool vec_ok = in_lds && ((gbase & 3) == 0);  // 16B global alignment
    const int  nvec   = total >> 2;

    // ---- issue the bulk staging copy FIRST, so it overlaps the count scan ----
    if (vec_ok) {
#if HAVE_ASYNC_LDS_LOAD
        gv4i_p gsrc = (gv4i_p)(scores + gbase);
        lv4i_p ldst = (lv4i_p)vals;
        for (int i = tid; i < nvec; i += BLOCK_THREADS)
            __builtin_amdgcn_global_load_async_to_lds_b128(gsrc + i, ldst + i, 0, 0);
#else
        const float4* __restrict__ src = (const float4*)(scores + gbase);
        for (int i = tid; i < nvec; i += BLOCK_THREADS)
            ((float4*)vals)[i] = src[i];
#endif
        for (int i = (nvec << 2) + tid; i < total; i += BLOCK_THREADS)
            vals[i] = scores[gbase + i];                 // scalar tail
    } else if (in_lds) {
        for (int i = tid; i < total; i += BLOCK_THREADS)
            vals[i] = scores[gbase + i];
    } else {
        // oversized block: warm the cache for the multi-pass fallback
        if (tid < 16) __builtin_prefetch(scores + gbase + tid * 32, 0, 0);
    }

    // ---- per-segment layout, overlapped with the copy above ----
    if (tid < SEGS_PER_BLOCK) {
        int seg = b * SEGS_PER_BLOCK + tid;
        s_cnt[tid] = (seg < num_seg) ? nbest[seg] : 0;
    }
    __syncthreads();
    if (tid == 0) {
        int acc = 0;
#pragma unroll
        for (int i = 0; i < SEGS_PER_BLOCK; ++i) { s_start[i] = acc; acc += s_cnt[i]; }
    }
#if HAVE_ASYNC_LDS_LOAD
    if (vec_ok) wait_asynccnt0();      // each wave drains its own async copies
#endif
    __syncthreads();

    const int wave = tid >> 5;
    const int lane = tid & 31;

    if (in_lds) {
        const int n  = s_cnt[wave];    // wave-uniform
        const int st = s_start[wave];  // wave-uniform
        if (n > 0) {
            float m = -3.402823466e38f;
            for (int i = lane; i < n; i += 32) m = fmaxf(m, vals[st + i]);
#pragma unroll
            for (int off = 16; off > 0; off >>= 1)
                m = fmaxf(m, __shfl_xor(m, off, 32));

            float sum = 0.0f;
            for (int i = lane; i < n; i += 32) {
                float e = __expf(vals[st + i] - m);
                vals[st + i] = e;
                sum += e;
            }
#pragma unroll
            for (int off = 16; off > 0; off >>= 1)
                sum += __shfl_xor(sum, off, 32);

            const float inv = 1.0f / sum;
            for (int i = lane; i < n; i += 32) vals[st + i] *= inv;
        }
        __syncthreads();

        if (vec_ok) {
            float4* __restrict__ dst = (float4*)(out + gbase);
            for (int i = tid; i < nvec; i += BLOCK_THREADS)
                dst[i] = ((const float4*)vals)[i];        // ds_b128 -> b128
            for (int i = (nvec << 2) + tid; i < total; i += BLOCK_THREADS)
                out[gbase + i] = vals[i];
        } else {
            for (int i = tid; i < total; i += BLOCK_THREADS)
                out[gbase + i] = vals[i];
        }
    } else {
        // ---- fallback: oversized ragged block, stream straight from global
        // (data stays L2-resident across passes; 192 MB global L2). ----
        int seg = b * SEGS_PER_BLOCK + wave;
        if (seg < num_seg) {
            const int n = s_cnt[wave];
            if (n > 0) {
                const long st = (long)gbase + s_start[wave];
                float m = -3.402823466e38f;
                for (int i = lane; i < n; i += 32) m = fmaxf(m, scores[st + i]);
#pragma unroll
                for (int off = 16; off > 0; off >>= 1)
                    m = fmaxf(m, __shfl_xor(m, off, 32));

                float sum = 0.0f;
                for (int i = lane; i < n; i += 32) sum += __expf(scores[st + i] - m);
#pragma unroll
                for (int off = 16; off > 0; off >>= 1)
                    sum += __shfl_xor(sum, off, 32);

                const float inv = 1.0f / sum;
                for (int i = lane; i < n; i += 32)
                    out[st + i] = __expf(scores[st + i] - m) * inv;
            }
        }
    }
}

// ---------------------------------------------------------------------------
// Host entry
// ---------------------------------------------------------------------------
extern "C" void kernel_launch(void* const* d_in, const int* in_sizes, int n_in,
                              void* d_out, int out_size, void* d_ws, size_t ws_size,
                              hipStream_t stream) {
    (void)n_in; (void)out_size; (void)ws_size;
    const float* scores = (const float*)d_in[0];
    const int*   nbest  = (const int*)d_in[1];
    float*       out    = (float*)d_out;

    const int num_seg = in_sizes[1];
    if (num_seg <= 0) return;

    const int ncb = (num_seg + SEGS_PER_BLOCK - 1) / SEGS_PER_BLOCK;
    int* block_bases = (int*)d_ws;   // (ncb+1) ints of scratch (~500 KB for 1M segs)

    const int gridA = (ncb + BLOCK_THREADS - 1) / BLOCK_THREADS;
    nbest_block_sums<<<gridA, BLOCK_THREADS, 0, stream>>>(nbest, num_seg,
                                                          block_bases, ncb);
    exclusive_scan_inplace<<<1, 1024, 0, stream>>>(block_bases, ncb);
    segmented_softmax<<<ncb, BLOCK_THREADS, 0, stream>>>(scores, nbest, out,
                                                         block_bases, num_seg);
}